// sparse_att_layer_68573447848480
// MI455X (gfx1250) — compile-verified
//
#include <hip/hip_runtime.h>

typedef __attribute__((ext_vector_type(2))) float v2f;
typedef __attribute__((ext_vector_type(4))) float v4f;
typedef __attribute__((ext_vector_type(8))) float v8f;

#define EMB 128

// Ordered-int trick for float atomic max (works for all finite floats, init -inf):
// nonnegative floats compare as signed ints; negative floats compare reversed as uints.
__device__ __forceinline__ void atomicMaxF32(float* addr, float v) {
  if (v >= 0.0f) {
    atomicMax((int*)addr, __float_as_int(v));
  } else {
    atomicMin((unsigned int*)addr, __float_as_uint(v));
  }
}

// ---------------------------------------------------------------------------
// 1) zero output accumulator, init row_max / row_sum
// ---------------------------------------------------------------------------
__global__ void k_init(float* __restrict__ out, float* __restrict__ row_max,
                       float* __restrict__ row_sum, int N, long long total) {
  long long i = (long long)blockIdx.x * blockDim.x + threadIdx.x;
  if (i < total) out[i] = 0.0f;
  if (i < N) {
    row_max[i] = -__builtin_inff();
    row_sum[i] = 0.0f;
  }
}

// ---------------------------------------------------------------------------
// 2) dual_transform = dual_layer @ conv_w + conv_b  via V_WMMA_F32_16X16X4_F32
//    One wave (32 lanes) computes 16 rows; K=256 accumulated in 64 WMMAs.
//    conv_w is broadcast across all 16 N-columns of B, so every column of D
//    holds the same dot product (robust to the B N-striping).
//    A-matrix 16x4 F32 layout (ISA 7.12.2): lanes 0-15 hold K=0 (VGPR0) and
//    K=1 (VGPR1); lanes 16-31 hold K=2 / K=3, all for M = lane&15.
// ---------------------------------------------------------------------------
__global__ void k_dual_transform(const float* __restrict__ dual_layer,
                                 const float* __restrict__ conv_w,
                                 const float* __restrict__ conv_b,
                                 float* __restrict__ dt, int R, int K) {
  const int lane = threadIdx.x;     // blockDim.x == 32, one wave
  const int base = blockIdx.x * 16;
  const int hi   = lane >> 4;       // lane half selects K pair
  const int m    = lane & 15;       // matrix row within tile
  int rel = base + m;
  if (rel >= R) rel = R - 1;        // clamp loads; out-of-range results unused
  const float* arow = dual_layer + (size_t)rel * (size_t)K;
  const int kx = hi ? 2 : 0;
  const int ky = hi ? 3 : 1;

  v8f c = {0.f, 0.f, 0.f, 0.f, 0.f, 0.f, 0.f, 0.f};
  for (int k0 = 0; k0 < K; k0 += 4) {
    v2f a, b;
    a.x = arow[k0 + kx];
    a.y = arow[k0 + ky];
    b.x = conv_w[k0 + kx];          // broadcast across all N columns
    b.y = conv_w[k0 + ky];
    c = __builtin_amdgcn_wmma_f32_16x16x4_f32(
        /*neg_a=*/false, a, /*neg_b=*/false, b,
        /*c_mod=*/(short)0, c, /*reuse_a=*/false, /*reuse_b=*/false);
  }

  // C/D layout: lane L, VGPR v holds D[M = v + 8*(L>=16)][N = L&15].
  // All N columns identical -> lanes 0 and 16 each own 8 distinct rows.
  const float bias = conv_b[0];
  if (m == 0) {
#pragma unroll
    for (int v = 0; v < 8; ++v) {
      int r = base + hi * 8 + v;
      if (r < R) dt[r] = c[v] + bias;
    }
  }
}

// ---------------------------------------------------------------------------
// 3) per-edge logit gather + leaky_relu + segment max (atomic)
// ---------------------------------------------------------------------------
__global__ void k_edge_logit_max(const int* __restrict__ rows,
                                 const int* __restrict__ rel_ids,
                                 const float* __restrict__ dt,
                                 float* __restrict__ lrelu_buf,
                                 float* __restrict__ row_max, int E) {
  int e = blockIdx.x * blockDim.x + threadIdx.x;
  if (e >= E) return;
  float t = dt[rel_ids[e]];
  float l = t >= 0.0f ? t : 0.01f * t;   // leaky_relu, slope 0.01
  lrelu_buf[e] = l;
  atomicMaxF32(&row_max[rows[e]], l);
}

// ---------------------------------------------------------------------------
// 4) per-edge exp(l - row_max) + segment sum (atomic)
// ---------------------------------------------------------------------------
__global__ void k_edge_exp_sum(const int* __restrict__ rows,
                               float* __restrict__ exbuf,
                               const float* __restrict__ row_max,
                               float* __restrict__ row_sum, int E) {
  int e = blockIdx.x * blockDim.x + threadIdx.x;
  if (e >= E) return;
  int r = rows[e];
  float x = __expf(exbuf[e] - row_max[r]);   // exponent <= 0, safe
  exbuf[e] = x;
  atomicAdd(&row_sum[r], x);
}

// ---------------------------------------------------------------------------
// 5) wave-per-edge weighted scatter-add: out[row] += coef * inlayer[col]
//    32 lanes x float4 = 128 floats; gathers mostly hit the 192MB L2
//    (inlayer is 51MB), atomics resolve in L2.
// ---------------------------------------------------------------------------
__global__ void k_aggregate(const int* __restrict__ rows,
                            const int* __restrict__ cols,
                            const float* __restrict__ exbuf,
                            const float* __restrict__ row_sum,
                            const float* __restrict__ inlayer,
                            float* __restrict__ out, int E) {
  long long tid = (long long)blockIdx.x * blockDim.x + threadIdx.x;
  int e    = (int)(tid >> 5);
  int lane = (int)(tid & 31);
  if (e >= E) return;
  int r    = rows[e];
  int cidx = cols[e];
  float coef = exbuf[e] / row_sum[r];        // row_sum >= 1 (max term = e^0)
  const v4f* src = (const v4f*)(inlayer + (size_t)cidx * EMB);
  v4f v = src[lane];                          // global_load_b128
  float* dst = out + (size_t)r * EMB + lane * 4;
  atomicAdd(dst + 0, coef * v.x);             // non-returning f32 atomics
  atomicAdd(dst + 1, coef * v.y);
  atomicAdd(dst + 2, coef * v.z);
  atomicAdd(dst + 3, coef * v.w);
}

// ---------------------------------------------------------------------------
// 6) final ReLU in place
// ---------------------------------------------------------------------------
__global__ void k_relu(float* __restrict__ out, long long total) {
  long long i = (long long)blockIdx.x * blockDim.x + threadIdx.x;
  if (i < total) {
    float v = out[i];
    out[i] = v > 0.0f ? v : 0.0f;
  }
}

extern "C" void kernel_launch(void* const* d_in, const int* in_sizes, int n_in,
                              void* d_out, int out_size, void* d_ws, size_t ws_size,
                              hipStream_t stream) {
  const float* inlayer    = (const float*)d_in[0];   // [N, 128]
  const float* dual_layer = (const float*)d_in[1];   // [R, 256]
  const int*   edge_index = (const int*)d_in[2];     // [2, E]
  const int*   rel_ids    = (const int*)d_in[3];     // [E]
  const float* conv_w     = (const float*)d_in[4];   // [256]
  const float* conv_b     = (const float*)d_in[5];   // [1]
  float* out = (float*)d_out;

  const int K = in_sizes[4];            // 2*EMB = 256
  const int R = in_sizes[1] / K;        // 1000
  const int E = in_sizes[3];            // 1,600,000
  const int N = in_sizes[0] / EMB;      // 100,000
  const int* rows = edge_index;         // edge_index[0, :]
  const int* cols = edge_index + E;     // edge_index[1, :]

  // workspace layout (floats): dt[R] | row_max[N] | row_sum[N] | exbuf[E]
  float* ws      = (float*)d_ws;
  float* dt      = ws;
  float* row_max = dt + ((R + 63) & ~63);
  float* row_sum = row_max + N;
  float* exbuf   = row_sum + N;

  const long long total = (long long)N * EMB;
  const int B = 256;

  k_init<<<(unsigned)((total + B - 1) / B), B, 0, stream>>>(out, row_max, row_sum, N, total);
  k_dual_transform<<<(R + 15) / 16, 32, 0, stream>>>(dual_layer, conv_w, conv_b, dt, R, K);
  k_edge_logit_max<<<(E + B - 1) / B, B, 0, stream>>>(rows, rel_ids, dt, exbuf, row_max, E);
  k_edge_exp_sum<<<(E + B - 1) / B, B, 0, stream>>>(rows, exbuf, row_max, row_sum, E);
  const long long aggT = (long long)E * 32;
  k_aggregate<<<(unsigned)((aggT + B - 1) / B), B, 0, stream>>>(rows, cols, exbuf, row_sum,
                                                               inlayer, out, E);
  k_relu<<<(unsigned)((total + B - 1) / B), B, 0, stream>>>(out, total);
}